// ContrastiveMSELoss_63067299774539
// MI455X (gfx1250) — compile-verified
//
#include <hip/hip_runtime.h>
#include <hip/hip_bf16.h>

// ---------------------------------------------------------------------------
// ContrastiveMSELoss for MI455X (gfx1250, wave32, WMMA).
//
// Algebra: with log_probs = cos - log(sum), ratings = 0.1 + sparse deltas:
//   loss_users   = -(1/N) [ S_cr - sum_i log(rowsum_i) * rowrat_i ]
//   loss_recipes = -(1/M) [ S_cr - sum_j log(colsum_j) * colrat_j ]
//   S_cr  = 0.1 * sum_ij cos + sum_winners (v - 0.1) * cos[u,i]
//   rowrat_i = 0.1*M + sum_winners_in_row (v - 0.1)   (col analogue)
// So we only need rowsum/colsum of exp(cos), total cos sum, and the sparse
// corrections -- never the 8192x8192 matrix itself.
// ---------------------------------------------------------------------------

#define NROWS 8192
#define NCOLS 8192
#define DIM   64
#define NBATCH 65536
#define EPS 1e-8f
#define FILLV 0.1f

typedef float v2f __attribute__((ext_vector_type(2)));
typedef float v8f __attribute__((ext_vector_type(8)));

// --------------------------- row norms -------------------------------------
__global__ void norms_kernel(const float* __restrict__ E, float* __restrict__ out, int n) {
  int i = blockIdx.x * blockDim.x + threadIdx.x;
  if (i >= n) return;
  const float* __restrict__ p = E + (size_t)i * DIM;
  float s = 0.f;
#pragma unroll
  for (int k = 0; k < DIM; k += 4) {
    float a = p[k], b = p[k + 1], c = p[k + 2], d = p[k + 3];
    s += a * a + b * b + c * c + d * d;
  }
  out[i] = sqrtf(s);
}

// --------------------------- main fused pass -------------------------------
// Grid: (NCOLS/16, NROWS/128), block = 256 threads = 8 waves.
// Each wave computes one 16x16 tile of cos via 16 chained fp32 WMMAs (K=4),
// then exp + rowsum/colsum/Sigma-cos accumulation.
__global__ __launch_bounds__(256) void cos_tile_kernel(
    const float* __restrict__ U, const float* __restrict__ R,
    const float* __restrict__ u_norm, const float* __restrict__ r_norm,
    float* __restrict__ rowsum, float* __restrict__ colsum,
    float* __restrict__ accum) {
  const int lane  = threadIdx.x & 31;
  const int wave  = threadIdx.x >> 5;
  const int row_base = (blockIdx.y * 8 + wave) * 16;
  const int col_base = blockIdx.x * 16;
  const int idx16 = lane & 15;   // M (for A) / N (for B,C,D)
  const int khalf = lane >> 4;   // which K pair within a K=4 chunk

  const float* __restrict__ arow = U + (size_t)(row_base + idx16) * DIM;
  const float* __restrict__ brow = R + (size_t)(col_base + idx16) * DIM;

  v8f acc = {};
#if __has_builtin(__builtin_amdgcn_wmma_f32_16x16x4_f32)
  // A 16x4 layout: lanes0-15 hold (K=kb,kb+1) for M=lane; lanes16-31 (K=kb+2,kb+3).
  // B 4x16 mirrored with N=lane&15.  D=64 -> 16 chained WMMAs.
#pragma unroll
  for (int k = 0; k < 16; ++k) {
    const int kb = 4 * k + 2 * khalf;
    v2f a = *(const v2f*)(arow + kb);
    v2f b = *(const v2f*)(brow + kb);
    acc = __builtin_amdgcn_wmma_f32_16x16x4_f32(false, a, false, b,
                                                (short)0, acc, false, false);
  }
#else
  // Scalar fallback matching the C/D layout (M = v + 8*khalf, N = idx16).
#pragma unroll
  for (int v = 0; v < 8; ++v) {
    const float* __restrict__ ur = U + (size_t)(row_base + v + 8 * khalf) * DIM;
    float s = 0.f;
#pragma unroll
    for (int k = 0; k < DIM; ++k) s += ur[k] * brow[k];
    acc[v] = s;
  }
#endif

  // Epilogue: C/D layout -> element (M = v + 8*(lane/16), N = lane%16).
  const int col = col_base + idx16;
  const float rn = r_norm[col];
  float cos_tot = 0.f;
  float col_acc = 0.f;
  float rowv[8];
#pragma unroll
  for (int v = 0; v < 8; ++v) {
    const int row = row_base + v + 8 * khalf;
    const float den = fmaxf(u_norm[row] * rn, EPS);
    const float c = acc[v] / den;
    const float e = __expf(c);
    cos_tot += c;
    col_acc += e;
    rowv[v] = e;
  }

  // Row sums: reduce across each 16-lane half (same M per half).
#pragma unroll
  for (int v = 0; v < 8; ++v) {
    float s = rowv[v];
    s += __shfl_xor(s, 1, 32);
    s += __shfl_xor(s, 2, 32);
    s += __shfl_xor(s, 4, 32);
    s += __shfl_xor(s, 8, 32);
    rowv[v] = s;
  }
  if (idx16 == 0) {
#pragma unroll
    for (int v = 0; v < 8; ++v)
      atomicAdd(&rowsum[row_base + v + 8 * khalf], rowv[v]);
  }

  // Col sums: per-lane 8 rows, add the other half (other 8 rows).
  col_acc += __shfl_xor(col_acc, 16, 32);
  if (lane < 16) atomicAdd(&colsum[col], col_acc);

  // Total cos sum: full wave reduce, one atomic.
  cos_tot += __shfl_xor(cos_tot, 1, 32);
  cos_tot += __shfl_xor(cos_tot, 2, 32);
  cos_tot += __shfl_xor(cos_tot, 4, 32);
  cos_tot += __shfl_xor(cos_tot, 8, 32);
  cos_tot += __shfl_xor(cos_tot, 16, 32);
  if (lane == 0) atomicAdd(&accum[0], cos_tot);
}

// --------------------- sparse scatter: last-write-wins ---------------------
__global__ void winner_kernel(const int* __restrict__ u_idx,
                              const int* __restrict__ i_idx,
                              int* __restrict__ winner, int n) {
  int b = blockIdx.x * blockDim.x + threadIdx.x;
  if (b >= n) return;
  size_t cell = (size_t)u_idx[b] * NCOLS + (size_t)i_idx[b];
  atomicMax(&winner[cell], b);  // max batch index == last write
}

__global__ void corr_kernel(const float* __restrict__ U, const float* __restrict__ R,
                            const float* __restrict__ u_norm, const float* __restrict__ r_norm,
                            const float* __restrict__ ratings, const float* __restrict__ cossim,
                            const int* __restrict__ u_idx, const int* __restrict__ i_idx,
                            const int* __restrict__ winner,
                            float* __restrict__ accum,
                            float* __restrict__ row_rat, float* __restrict__ col_rat, int n) {
  int b = blockIdx.x * blockDim.x + threadIdx.x;
  // MSE over all B entries (no dedup), wave-reduced before atomic.
  float msep = 0.f;
  if (b < n) {
    float d = ratings[b] - cossim[b];
    msep = d * d;
  }
  msep += __shfl_xor(msep, 1, 32);
  msep += __shfl_xor(msep, 2, 32);
  msep += __shfl_xor(msep, 4, 32);
  msep += __shfl_xor(msep, 8, 32);
  msep += __shfl_xor(msep, 16, 32);
  if ((threadIdx.x & 31) == 0) atomicAdd(&accum[2], msep);
  if (b >= n) return;

  const int u = u_idx[b], i = i_idx[b];
  bool win = winner ? (winner[(size_t)u * NCOLS + (size_t)i] == b) : true;
  if (!win) return;

  const float* __restrict__ ur = U + (size_t)u * DIM;
  const float* __restrict__ rr = R + (size_t)i * DIM;
  float s = 0.f;
#pragma unroll
  for (int k = 0; k < DIM; ++k) s += ur[k] * rr[k];
  const float c = s / fmaxf(u_norm[u] * r_norm[i], EPS);
  const float w = ratings[b] - FILLV;
  atomicAdd(&accum[1], w * c);
  atomicAdd(&row_rat[u], w);
  atomicAdd(&col_rat[i], w);
}

// ------------------------------ finalize -----------------------------------
__global__ __launch_bounds__(1024) void finalize_kernel(
    const float* __restrict__ rowsum, const float* __restrict__ colsum,
    const float* __restrict__ row_rat, const float* __restrict__ col_rat,
    const float* __restrict__ accum, float* __restrict__ out) {
  __shared__ float sa[32], sb[32];
  float a = 0.f, b = 0.f;
  const float base = FILLV * (float)NCOLS;  // background row/col rating mass
  for (int i = threadIdx.x; i < NROWS; i += 1024) {
    a += logf(rowsum[i]) * (base + row_rat[i]);
    b += logf(colsum[i]) * (base + col_rat[i]);
  }
#pragma unroll
  for (int m = 1; m < 32; m <<= 1) {
    a += __shfl_xor(a, m, 32);
    b += __shfl_xor(b, m, 32);
  }
  const int wave = threadIdx.x >> 5, lane = threadIdx.x & 31;
  if (lane == 0) { sa[wave] = a; sb[wave] = b; }
  __syncthreads();
  if (threadIdx.x == 0) {
    float A = 0.f, Bv = 0.f;
    for (int w = 0; w < 32; ++w) { A += sa[w]; Bv += sb[w]; }
    const float S_cr = FILLV * accum[0] + accum[1];
    const float inv  = 1.0f / (float)NROWS;
    const float loss_users   = -(S_cr - A)  * inv;
    const float loss_recipes = -(S_cr - Bv) * inv;
    const float contrastive  = 0.5f * (loss_users + loss_recipes);
    const float mse = accum[2] * (1.0f / (float)NBATCH);
    out[0] = 0.5f * contrastive + 0.5f * mse;
  }
}

// ------------------------------ launcher -----------------------------------
extern "C" void kernel_launch(void* const* d_in, const int* in_sizes, int n_in,
                              void* d_out, int out_size, void* d_ws, size_t ws_size,
                              hipStream_t stream) {
  const float* U       = (const float*)d_in[0];
  const float* R       = (const float*)d_in[1];
  const float* ratings = (const float*)d_in[2];
  const float* cossim  = (const float*)d_in[3];
  const int*   u_idx   = (const int*)d_in[4];
  const int*   i_idx   = (const int*)d_in[5];
  float* out = (float*)d_out;

  // Workspace layout (floats):
  // accum[8] | rowsum[N] | colsum[M] | row_rat[N] | col_rat[M] | u_norm[N] | r_norm[M]
  float* ws      = (float*)d_ws;
  float* accum   = ws;
  float* rowsum  = accum + 8;
  float* colsum  = rowsum + NROWS;
  float* row_rat = colsum + NCOLS;
  float* col_rat = row_rat + NROWS;
  float* u_norm  = col_rat + NCOLS;
  float* r_norm  = u_norm + NROWS;
  size_t small_bytes = (size_t)(8 + 4 * NROWS + 2 * NCOLS) * sizeof(float);

  // Optional dedup (last-write-wins) array if workspace is large enough.
  size_t winner_off   = (small_bytes + 255) & ~(size_t)255;
  size_t winner_bytes = (size_t)NROWS * NCOLS * sizeof(int);
  int* winner = nullptr;
  if (ws_size >= winner_off + winner_bytes) {
    winner = (int*)((char*)d_ws + winner_off);
    hipMemsetAsync(winner, 0xFF, winner_bytes, stream);  // -1
  }
  hipMemsetAsync(d_ws, 0, small_bytes, stream);

  norms_kernel<<<NROWS / 256, 256, 0, stream>>>(U, u_norm, NROWS);
  norms_kernel<<<NCOLS / 256, 256, 0, stream>>>(R, r_norm, NCOLS);

  cos_tile_kernel<<<dim3(NCOLS / 16, NROWS / 128), 256, 0, stream>>>(
      U, R, u_norm, r_norm, rowsum, colsum, accum);

  if (winner)
    winner_kernel<<<NBATCH / 256, 256, 0, stream>>>(u_idx, i_idx, winner, NBATCH);

  corr_kernel<<<NBATCH / 256, 256, 0, stream>>>(U, R, u_norm, r_norm, ratings, cossim,
                                                u_idx, i_idx, winner, accum,
                                                row_rat, col_rat, NBATCH);

  finalize_kernel<<<1, 1024, 0, stream>>>(rowsum, colsum, row_rat, col_rat, accum, out);

  (void)in_sizes; (void)n_in; (void)out_size;
}